// GRACE_51230369906743
// MI455X (gfx1250) — compile-verified
//
#include <hip/hip_runtime.h>
#include <hip/hip_bf16.h>
#include <math.h>

typedef __attribute__((ext_vector_type(2))) float v2f;
typedef __attribute__((ext_vector_type(8))) float v8f;

#define BN_EPS 1e-5f

// ---------------------------------------------------------------------------
// GEMM1: h1[N,64] = x[N,8] @ W1[8,64]   (V_WMMA_F32_16X16X4_F32, K=8 -> 2 steps)
// One wave per 16-node tile; 4 output tiles of 16x16 per wave.
// W1 (2KB) staged in LDS and broadcast to all waves of the block.
// ---------------------------------------------------------------------------
__global__ __launch_bounds__(64) void gcn_gemm1(const float* __restrict__ x,
                                                const float* __restrict__ W1,
                                                float* __restrict__ h1,
                                                int n_nodes, int n_tiles) {
    __shared__ float sW[8 * 64];
#pragma unroll
    for (int i = threadIdx.x; i < 8 * 64; i += 64) sW[i] = W1[i];
    __syncthreads();

    int tile = blockIdx.x * 2 + (threadIdx.x >> 5);
    if (tile >= n_tiles) return;                 // wave-uniform in practice
    int lane = threadIdx.x & 31;
    int m    = lane & 15;                        // A row / B,C col within tile
    int hb   = lane >> 4;                        // lane half
    int kb   = hb * 2;                           // K offset within 4-wide step

    int row    = tile * 16 + m;
    int rclamp = row < n_nodes ? row : n_nodes - 1;   // keep EXEC all-1s

    // A fragments for the two K-steps (K=0..3 and K=4..7)
    v2f a0 = *(const v2f*)&x[(size_t)rclamp * 8 + 0 + kb];
    v2f a1 = *(const v2f*)&x[(size_t)rclamp * 8 + 4 + kb];

    // wave-uniform full-tile test, forced scalar
    int  t0   = __builtin_amdgcn_readfirstlane(tile);
    bool full = (t0 * 16 + 16) <= n_nodes;

#pragma unroll
    for (int nt = 0; nt < 4; ++nt) {
        int col = nt * 16 + m;
        v2f b0 = { sW[(0 + kb) * 64 + col], sW[(1 + kb) * 64 + col] };
        v2f b1 = { sW[(4 + kb) * 64 + col], sW[(5 + kb) * 64 + col] };
        v8f c = {};
        c = __builtin_amdgcn_wmma_f32_16x16x4_f32(false, a0, false, b0,
                                                  (short)0, c, false, false);
        c = __builtin_amdgcn_wmma_f32_16x16x4_f32(false, a1, false, b1,
                                                  (short)0, c, false, false);
        int rbase = tile * 16 + hb * 8;
        if (full) {
#pragma unroll
            for (int v = 0; v < 8; ++v)
                h1[(size_t)(rbase + v) * 64 + col] = c[v];
        } else {
#pragma unroll
            for (int v = 0; v < 8; ++v)
                if (rbase + v < n_nodes) h1[(size_t)(rbase + v) * 64 + col] = c[v];
        }
    }
}

// ---------------------------------------------------------------------------
// GEMM2: h2[N,32] = hbn[N,64] @ W2[64,32]   (K=64 -> 16 WMMA steps)
// One wave per 16-node tile; 2 output tiles of 16x16 per wave.
// W2 (8KB) staged in LDS.
// ---------------------------------------------------------------------------
__global__ __launch_bounds__(64) void gcn_gemm2(const float* __restrict__ hbn,
                                                const float* __restrict__ W2,
                                                float* __restrict__ h2,
                                                int n_nodes, int n_tiles) {
    __shared__ float sW[64 * 32];
#pragma unroll
    for (int i = threadIdx.x; i < 64 * 32; i += 64) sW[i] = W2[i];
    __syncthreads();

    int tile = blockIdx.x * 2 + (threadIdx.x >> 5);
    if (tile >= n_tiles) return;
    int lane = threadIdx.x & 31;
    int m  = lane & 15;
    int hb = lane >> 4;
    int kb = hb * 2;

    int row    = tile * 16 + m;
    int rclamp = row < n_nodes ? row : n_nodes - 1;

    v2f a[16];
#pragma unroll
    for (int k = 0; k < 16; ++k)
        a[k] = *(const v2f*)&hbn[(size_t)rclamp * 64 + k * 4 + kb];

    int  t0   = __builtin_amdgcn_readfirstlane(tile);
    bool full = (t0 * 16 + 16) <= n_nodes;

#pragma unroll
    for (int nt = 0; nt < 2; ++nt) {
        int col = nt * 16 + m;
        v8f c = {};
#pragma unroll
        for (int k = 0; k < 16; ++k) {
            v2f b = { sW[(k * 4 + kb) * 32 + col],
                      sW[(k * 4 + kb + 1) * 32 + col] };
            c = __builtin_amdgcn_wmma_f32_16x16x4_f32(false, a[k], false, b,
                                                      (short)0, c, false, false);
        }
        int rbase = tile * 16 + hb * 8;
        if (full) {
#pragma unroll
            for (int v = 0; v < 8; ++v)
                h2[(size_t)(rbase + v) * 32 + col] = c[v];
        } else {
#pragma unroll
            for (int v = 0; v < 8; ++v)
                if (rbase + v < n_nodes) h2[(size_t)(rbase + v) * 32 + col] = c[v];
        }
    }
}

// ---------------------------------------------------------------------------
// Degree (f32 atomic histogram over dst), then dinv = rsqrt(deg+1) in place.
// ---------------------------------------------------------------------------
__global__ void degree_kernel(const int* __restrict__ dst, float* deg, int n_edges) {
    int e = blockIdx.x * blockDim.x + threadIdx.x;
    if (e < n_edges) atomicAdd(&deg[dst[e]], 1.0f);
}

__global__ void dinv_kernel(float* deg, int n_nodes) {
    int i = blockIdx.x * blockDim.x + threadIdx.x;
    if (i < n_nodes) deg[i] = rsqrtf(deg[i] + 1.0f);
}

// ---------------------------------------------------------------------------
// Edge scatter: agg[dst] += h[src] * dinv[src]*dinv[dst]
// One thread per (edge, 4-channel group): float4 gather + 4 f32 atomics.
// h / dinv / edge index are L2-resident (192MB L2); this is atomic-throughput
// bound, not HBM bound.
// ---------------------------------------------------------------------------
template <int CH>
__global__ void scatter_kernel(const float* __restrict__ h,
                               const int* __restrict__ src,
                               const int* __restrict__ dst,
                               const float* __restrict__ dinv,
                               float* agg, int n_edges) {
    constexpr int G  = CH / 4;
    constexpr int SH = (CH == 64) ? 4 : 3;
    long long t = (long long)blockIdx.x * blockDim.x + threadIdx.x;
    if (t >= (long long)n_edges * G) return;
    int e = (int)(t >> SH);
    int g = (int)(t & (G - 1));
    int s = src[e];
    int d = dst[e];
    float nrm = dinv[s] * dinv[d];
    const float4 hv = *(const float4*)&h[(size_t)s * CH + g * 4];
    float* ap = &agg[(size_t)d * CH + g * 4];
    atomicAdd(ap + 0, hv.x * nrm);
    atomicAdd(ap + 1, hv.y * nrm);
    atomicAdd(ap + 2, hv.z * nrm);
    atomicAdd(ap + 3, hv.w * nrm);
}

// ---------------------------------------------------------------------------
// Layer-1 epilogue: self-loop + bias + BatchNorm(inference) + ReLU.
// hbn may alias agg1 (pure elementwise, same index).
// ---------------------------------------------------------------------------
__global__ void finalize1_kernel(const float* agg1, const float* __restrict__ h1,
                                 const float* __restrict__ dinv,
                                 const float* __restrict__ b1,
                                 const float* __restrict__ gamma,
                                 const float* __restrict__ beta,
                                 const float* __restrict__ rmean,
                                 const float* __restrict__ rvar,
                                 float* hbn, int n_nodes) {
    int t = blockIdx.x * blockDim.x + threadIdx.x;
    if (t >= n_nodes * 64) return;
    int i = t >> 6;
    int c = t & 63;
    float di = dinv[i];
    float v  = agg1[t] + h1[t] * (di * di) + b1[c];
    v = (v - rmean[c]) * rsqrtf(rvar[c] + BN_EPS) * gamma[c] + beta[c];
    hbn[t] = v > 0.0f ? v : 0.0f;
}

// Layer-2 epilogue: self-loop + bias -> output
__global__ void finalize2_kernel(const float* __restrict__ agg2,
                                 const float* __restrict__ h2,
                                 const float* __restrict__ dinv,
                                 const float* __restrict__ b2,
                                 float* __restrict__ out, int n_nodes) {
    int t = blockIdx.x * blockDim.x + threadIdx.x;
    if (t >= n_nodes * 32) return;
    int i = t >> 5;
    int c = t & 31;
    float di = dinv[i];
    out[t] = agg2[t] + h2[t] * (di * di) + b2[c];
}

// ---------------------------------------------------------------------------
extern "C" void kernel_launch(void* const* d_in, const int* in_sizes, int n_in,
                              void* d_out, int out_size, void* d_ws, size_t ws_size,
                              hipStream_t stream) {
    const float* x     = (const float*)d_in[0];
    const int*   eidx  = (const int*)  d_in[1];
    const float* W1    = (const float*)d_in[2];
    const float* b1    = (const float*)d_in[3];
    const float* gamma = (const float*)d_in[4];
    const float* beta  = (const float*)d_in[5];
    const float* rmean = (const float*)d_in[6];
    const float* rvar  = (const float*)d_in[7];
    const float* W2    = (const float*)d_in[8];
    const float* b2    = (const float*)d_in[9];
    float* out = (float*)d_out;

    const int n_nodes = in_sizes[0] / 8;
    const int n_edges = in_sizes[1] / 2;
    const int* src = eidx;
    const int* dst = eidx + n_edges;

    // Workspace carve (aligned to 256B). deg buffer becomes dinv in place;
    // agg1 becomes hbn in place.  Total ~77MB.
    char* p = (char*)d_ws;
    auto carve = [&](size_t bytes) -> char* {
        char* q = p;
        p += (bytes + 255) & ~(size_t)255;
        return q;
    };
    float* deg  = (float*)carve((size_t)n_nodes * 4);        // -> dinv
    float* h1   = (float*)carve((size_t)n_nodes * 64 * 4);
    float* agg1 = (float*)carve((size_t)n_nodes * 64 * 4);   // -> hbn
    float* h2   = (float*)carve((size_t)n_nodes * 32 * 4);
    float* agg2 = (float*)carve((size_t)n_nodes * 32 * 4);

    hipMemsetAsync(deg,  0, (size_t)n_nodes * 4,      stream);
    hipMemsetAsync(agg1, 0, (size_t)n_nodes * 64 * 4, stream);
    hipMemsetAsync(agg2, 0, (size_t)n_nodes * 32 * 4, stream);

    // degrees + dinv
    degree_kernel<<<(n_edges + 255) / 256, 256, 0, stream>>>(dst, deg, n_edges);
    dinv_kernel<<<(n_nodes + 255) / 256, 256, 0, stream>>>(deg, n_nodes);

    // layer 1
    const int n_tiles     = (n_nodes + 15) / 16;
    const int gemm_blocks = (n_tiles + 1) / 2;   // 2 waves (tiles) per 64-thread block
    gcn_gemm1<<<gemm_blocks, 64, 0, stream>>>(x, W1, h1, n_nodes, n_tiles);

    long long s1 = (long long)n_edges * 16;
    scatter_kernel<64><<<(int)((s1 + 255) / 256), 256, 0, stream>>>(
        h1, src, dst, deg, agg1, n_edges);

    finalize1_kernel<<<(n_nodes * 64 + 255) / 256, 256, 0, stream>>>(
        agg1, h1, deg, b1, gamma, beta, rmean, rvar, agg1, n_nodes);

    // layer 2
    gcn_gemm2<<<gemm_blocks, 64, 0, stream>>>(agg1, W2, h2, n_nodes, n_tiles);

    long long s2 = (long long)n_edges * 8;
    scatter_kernel<32><<<(int)((s2 + 255) / 256), 256, 0, stream>>>(
        h2, src, dst, deg, agg2, n_edges);

    finalize2_kernel<<<(n_nodes * 32 + 255) / 256, 256, 0, stream>>>(
        agg2, h2, deg, b2, out, n_nodes);
}